// GPR_sparse_32401233281228
// MI455X (gfx1250) — compile-verified
//
#include <hip/hip_runtime.h>

typedef __attribute__((ext_vector_type(2))) float v2f;
typedef __attribute__((ext_vector_type(8))) float v8f;

#define D 128
#define NLAYERS 4
#define LDS_STRIDE 132   // 132*4B = 528B row stride: 16B-aligned, bank stride 4 -> conflict-free

// ---------------------------------------------------------------------------
// init: hidden = temp[0]*x ; cur = x ; y = 0     (float4 granularity)
// ---------------------------------------------------------------------------
__global__ __launch_bounds__(256) void gpr_init_kernel(
    const float* __restrict__ x, const float* __restrict__ temp,
    float* __restrict__ hidden, float* __restrict__ cur, float* __restrict__ y,
    int n4) {
  int i = blockIdx.x * 256 + threadIdx.x;
  if (i >= n4) return;
  float t0 = temp[0];
  float4 v = ((const float4*)x)[i];
  float4 hv; hv.x = v.x * t0; hv.y = v.y * t0; hv.z = v.z * t0; hv.w = v.w * t0;
  ((float4*)hidden)[i] = hv;
  ((float4*)cur)[i] = v;
  float4 z; z.x = 0.f; z.y = 0.f; z.z = 0.f; z.w = 0.f;
  ((float4*)y)[i] = z;
}

// ---------------------------------------------------------------------------
// GEMM: H = A @ Wm + bias, A:[N,128], Wm:[128,128], via V_WMMA_F32_16X16X4_F32
// block = 256 threads = 8 waves; block computes rows [16*bid, 16*bid+16) x all 128 cols
// wave w owns the 16-col tile [16w, 16w+16); K-loop = 32 WMMAs of K=4
// ---------------------------------------------------------------------------
__global__ __launch_bounds__(256) void gpr_gemm_kernel(
    const float* __restrict__ A, const float* __restrict__ Wm,
    const float* __restrict__ bias, float* __restrict__ H) {
  __shared__ float As[16 * LDS_STRIDE];

  const int tid  = threadIdx.x;
  const int row0 = blockIdx.x * 16;

  // Stage the 16x128 A tile into LDS (512 float4 loads, 2 per thread, uniform)
  for (int i = tid; i < 512; i += 256) {
    int r  = i >> 5;      // 0..15
    int c4 = i & 31;      // 0..31 (float4 index)
    float4 v = *(const float4*)(A + (size_t)(row0 + r) * D + c4 * 4);
    *(float4*)(&As[r * LDS_STRIDE + c4 * 4]) = v;
  }
  __syncthreads();

  const int wave  = tid >> 5;            // 0..7 -> column tile
  const int lane  = tid & 31;
  const int n     = lane & 15;           // col within tile (B/C/D), row within tile (A)
  const int kbase = (lane >> 4) << 1;    // 0 for lanes 0-15, 2 for lanes 16-31
  const int col   = wave * 16 + n;

  // Preload all B fragments (W is L2-resident & reused by all 3125 blocks)
  v2f Bfrag[32];
#pragma unroll
  for (int t = 0; t < 32; ++t) {
    int k = 4 * t + kbase;
    Bfrag[t].x = Wm[(size_t)k * D + col];
    Bfrag[t].y = Wm[(size_t)(k + 1) * D + col];
  }

  // Bias folded into accumulator init (C/D col = n for every VGPR)
  float bv = bias[col];
  v8f acc = {bv, bv, bv, bv, bv, bv, bv, bv};

  const float* arow = &As[n * LDS_STRIDE];
#pragma unroll
  for (int t = 0; t < 32; ++t) {
    v2f a;
    a.x = arow[4 * t + kbase];
    a.y = arow[4 * t + kbase + 1];
    // (neg_a, A, neg_b, B, c_mod, C, reuse_a, reuse_b)
    acc = __builtin_amdgcn_wmma_f32_16x16x4_f32(
        false, a, false, Bfrag[t], (short)0, acc, false, false);
  }

  // C/D layout: VGPR r -> M = r + 8*(lane/16), N = lane%16
  const int mhalf = (lane >> 4) * 8;
#pragma unroll
  for (int r = 0; r < 8; ++r) {
    H[(size_t)(row0 + r + mhalf) * D + col] = acc[r];
  }
}

// ---------------------------------------------------------------------------
// Edge scatter: one wave32 per edge, 4 channels per lane (float4 = 512B row / wave)
// y[dst] += h[src] * w  via non-returning f32 atomics (STOREcnt, L2-resident)
// ---------------------------------------------------------------------------
__global__ __launch_bounds__(256) void gpr_scatter_kernel(
    const float* __restrict__ H, const float* __restrict__ w,
    const int* __restrict__ src, const int* __restrict__ dst,
    float* __restrict__ y, int E) {
  unsigned gid = blockIdx.x * 256u + threadIdx.x;
  int e    = (int)(gid >> 5);
  int lane = (int)(gid & 31u);
  if (e >= E) return;
  int s   = src[e];
  int d   = dst[e];
  float ww = w[e];
  float4 m = *(const float4*)(H + (size_t)s * D + lane * 4);
  float* yp = y + (size_t)d * D + lane * 4;
  unsafeAtomicAdd(yp + 0, m.x * ww);
  unsafeAtomicAdd(yp + 1, m.y * ww);
  unsafeAtomicAdd(yp + 2, m.z * ww);
  unsafeAtomicAdd(yp + 3, m.w * ww);
}

// ---------------------------------------------------------------------------
// combine: r = relu(y); cur = r; hidden += temp[layer+1]*r; y = 0 (ready for next layer)
// ---------------------------------------------------------------------------
__global__ __launch_bounds__(256) void gpr_combine_kernel(
    float* __restrict__ y, float* __restrict__ cur, float* __restrict__ hidden,
    const float* __restrict__ temp, int layer, int n4) {
  int i = blockIdx.x * 256 + threadIdx.x;
  if (i >= n4) return;
  float t = temp[layer + 1];
  float4 v = ((float4*)y)[i];
  float4 r;
  r.x = v.x > 0.f ? v.x : 0.f;
  r.y = v.y > 0.f ? v.y : 0.f;
  r.z = v.z > 0.f ? v.z : 0.f;
  r.w = v.w > 0.f ? v.w : 0.f;
  ((float4*)cur)[i] = r;
  float4 h = ((float4*)hidden)[i];
  h.x += r.x * t; h.y += r.y * t; h.z += r.z * t; h.w += r.w * t;
  ((float4*)hidden)[i] = h;
  float4 z; z.x = 0.f; z.y = 0.f; z.z = 0.f; z.w = 0.f;
  ((float4*)y)[i] = z;
}

// ---------------------------------------------------------------------------
extern "C" void kernel_launch(void* const* d_in, const int* in_sizes, int n_in,
                              void* d_out, int out_size, void* d_ws, size_t ws_size,
                              hipStream_t stream) {
  const float* x    = (const float*)d_in[0];
  const float* w    = (const float*)d_in[1];
  const int*   src  = (const int*)d_in[2];
  const int*   dst  = (const int*)d_in[3];
  const float* Wm   = (const float*)d_in[4];   // [L,128,128]
  const float* bias = (const float*)d_in[5];   // [L,128]
  const float* temp = (const float*)d_in[6];   // [L+1]
  float* hidden     = (float*)d_out;

  const int E = in_sizes[1];
  const int N = in_sizes[0] / D;               // 50000

  float* cur = (float*)d_ws;
  float* h   = cur + (size_t)N * D;
  float* y   = h   + (size_t)N * D;

  const int n4 = N * D / 4;
  const int eb = (n4 + 255) / 256;

  gpr_init_kernel<<<eb, 256, 0, stream>>>(x, temp, hidden, cur, y, n4);

  for (int l = 0; l < NLAYERS; ++l) {
    gpr_gemm_kernel<<<N / 16, 256, 0, stream>>>(cur, Wm + (size_t)l * D * D,
                                                bias + (size_t)l * D, h);
    int sblocks = (int)(((long long)E * 32 + 255) / 256);
    gpr_scatter_kernel<<<sblocks, 256, 0, stream>>>(h, w, src, dst, y, E);
    gpr_combine_kernel<<<eb, 256, 0, stream>>>(y, cur, hidden, temp, l, n4);
  }
}